// RuleScorer_54374285968080
// MI455X (gfx1250) — compile-verified
//
#include <hip/hip_runtime.h>
#include <hip/hip_bf16.h>
#include <math.h>

// Problem constants (match reference)
#define BB 4
#define NN 48
#define RR 16
#define TT 33
#define NC 80
#define LL 3
#define NT 16
#define NI 5
#define NEGF (-1000.0f)

#define N2 (NN * NN)            // 2304 elements per 48x48 matrix
#define MROWS (BB * N2)         // 9216 rows of s
typedef __attribute__((ext_vector_type(2))) float v2f;
typedef __attribute__((ext_vector_type(8))) float v8f;

// ---------------------------------------------------------------------------
// Phase 1: tropical chain DP per (batch, chain).
// grid = (NC, BB), block = 256 threads (8 waves).
// H0,H1,H2 gathered into LDS; two max-plus 48x48x48 matmuls in LDS; exp();
// result written to ws as s[row=b*2304+x*48+y][c] (chain index contiguous).
// ---------------------------------------------------------------------------
__global__ __launch_bounds__(256)
void path_dp_kernel(const float* __restrict__ transitions,
                    const int* __restrict__ rules,
                    float* __restrict__ s_out) {
    __shared__ float H0[N2];
    __shared__ float H1[N2];
    __shared__ float H2[N2];
    __shared__ float M1[N2];

    const int c = blockIdx.x;           // chain
    const int b = blockIdx.y;           // batch
    const int tid = threadIdx.x;

    const int t0 = rules[c * LL + 0];
    const int t1 = rules[c * LL + 1];
    const int t2 = rules[c * LL + 2];

    const float* tb = transitions + (size_t)b * N2 * TT;

    // Gather the three hop matrices for this (b,c) into LDS.
    #pragma unroll
    for (int i = 0; i < N2 / 256; ++i) {
        int e = tid + i * 256;
        const float* p = tb + (size_t)e * TT;
        H0[e] = p[t0];
        H1[e] = p[t1];
        H2[e] = p[t2];
    }
    __syncthreads();

    // M1 = H0 (x) H1   (max-plus matmul)
    #pragma unroll
    for (int i = 0; i < N2 / 256; ++i) {
        int e = tid + i * 256;
        int x = e / NN, y = e % NN;
        const float* a = &H0[x * NN];
        float m = -INFINITY;
        #pragma unroll 8
        for (int k = 0; k < NN; ++k)
            m = fmaxf(m, a[k] + H1[k * NN + y]);
        M1[e] = m;
    }
    __syncthreads();

    // S = M1 (x) H2 ; s_out = exp(S)
    #pragma unroll
    for (int i = 0; i < N2 / 256; ++i) {
        int e = tid + i * 256;
        int x = e / NN, y = e % NN;
        const float* a = &M1[x * NN];
        float m = -INFINITY;
        #pragma unroll 8
        for (int k = 0; k < NN; ++k)
            m = fmaxf(m, a[k] + H2[k * NN + y]);
        s_out[(size_t)(b * N2 + e) * NC + c] = __expf(m);
    }
}

// ---------------------------------------------------------------------------
// Phase 2: comb = s[9216x80] x W[80x16] via V_WMMA_F32_16X16X4_F32 (f32 exact),
// where W is block-diagonal: W[c][t] = (c/5==t) ? weights_flat[c] : 0.
// Then +bias, mask with type_mask==0 else -1000.
// block = 128 threads (4 waves); each wave owns one 16-row tile.
// grid = MROWS / (16*4) = 144 blocks.
// ---------------------------------------------------------------------------
__global__ __launch_bounds__(128)
void combine_wmma_kernel(const float* __restrict__ s_in,
                         const float* __restrict__ weights,   // [16,5] flat
                         const float* __restrict__ biases,    // [16]
                         const int* __restrict__ type_mask,   // [MROWS,16] flat
                         float* __restrict__ out) {           // [MROWS,16] flat
    __shared__ float Wlds[NC * NT];        // 80 x 16 block-diagonal weights
    __shared__ float Slds[4][16 * NC];     // per-wave 16 x 80 tile of s

    const int tid  = threadIdx.x;
    const int wave = tid >> 5;
    const int lane = tid & 31;

    // Build block-diagonal W: W[c][t] = (c/NI == t) ? weights[c] : 0
    for (int i = tid; i < NC * NT; i += 128) {
        int c = i / NT, t = i % NT;
        Wlds[i] = (c / NI == t) ? weights[c] : 0.0f;
    }

    // Stage this wave's 16x80 tile of s (rows are contiguous: coalesced).
    const int tile = blockIdx.x * 4 + wave;
    const int m0 = tile * 16;
    const float* srow = s_in + (size_t)m0 * NC;
    #pragma unroll
    for (int i = lane; i < 16 * NC; i += 32)
        Slds[wave][i] = srow[i];

    __syncthreads();

    // Fragment addressing (ISA 7.12.2, 32-bit f32 layouts):
    //  A 16x4: lane%16 = M; lane-half selects K pair {0,1} vs {2,3}; vgpr = K lsb
    //  B 4x16: lane%16 = N; lane-half selects K pair; vgpr = K lsb
    const int mn   = lane & 15;
    const int khalf = (lane >> 4) << 1;     // 0 or 2

    v8f acc = {};
    #pragma unroll
    for (int q = 0; q < NC / 4; ++q) {
        const int kbase = 4 * q + khalf;
        v2f a, bfrag;
        a.x = Slds[wave][mn * NC + kbase];
        a.y = Slds[wave][mn * NC + kbase + 1];
        bfrag.x = Wlds[kbase * NT + mn];
        bfrag.y = Wlds[(kbase + 1) * NT + mn];
        acc = __builtin_amdgcn_wmma_f32_16x16x4_f32(
            /*neg_a=*/false, a, /*neg_b=*/false, bfrag,
            /*c_mod=*/(short)0, acc, /*reuse_a=*/false, /*reuse_b=*/false);
    }

    // Epilogue: D layout — vgpr r: lanes0-15 -> M=r, lanes16-31 -> M=8+r; N=lane%16
    const int t = mn;
    const float bias = biases[t];
    const int mbase = m0 + ((lane >> 4) << 3);   // +8 for upper lane half
    #pragma unroll
    for (int r = 0; r < 8; ++r) {
        const int row = mbase + r;
        const int idx = row * RR + t;
        const float v = acc[r] + bias;
        out[idx] = (type_mask[idx] == 0) ? v : NEGF;
    }
}

// ---------------------------------------------------------------------------
extern "C" void kernel_launch(void* const* d_in, const int* in_sizes, int n_in,
                              void* d_out, int out_size, void* d_ws, size_t ws_size,
                              hipStream_t stream) {
    const float* transitions = (const float*)d_in[0];
    const int*   type_mask   = (const int*)d_in[1];
    const int*   rules       = (const int*)d_in[2];
    const float* weights     = (const float*)d_in[3];
    const float* biases      = (const float*)d_in[4];
    float* out = (float*)d_out;
    float* s_ws = (float*)d_ws;   // needs MROWS*NC*4 = ~2.95 MB

    dim3 g1(NC, BB);
    path_dp_kernel<<<g1, 256, 0, stream>>>(transitions, rules, s_ws);

    dim3 g2(MROWS / (16 * 4));    // 144 blocks, 4 wave-tiles each
    combine_wmma_kernel<<<g2, 128, 0, stream>>>(s_ws, weights, biases,
                                                type_mask, out);
}